// ChamferDistance_4750233829481
// MI455X (gfx1250) — compile-verified
//
#include <hip/hip_runtime.h>
#include <math.h>

typedef __attribute__((ext_vector_type(2))) float v2f;
typedef __attribute__((ext_vector_type(8))) float v8f;

// Pass 0: rows = gt,   cols = pred  -> dist1[b, i] = min_j ||pred_j - gt_i||
// Pass 1: rows = pred, cols = gt    -> dist2[b, j] = min_i ||pred_j - gt_i||
//
// One wave (32 lanes) owns one 16-row tile and sweeps all N/16 column tiles.
// sq_dist tile computed with V_WMMA_F32_16X16X4_F32:
//   A row i   = ( x_i,  y_i,  z_i, ||r_i||^2 )      (16x4)
//   B col j   = (-2x_j,-2y_j,-2z_j, 1 )             (4x16)
//   D[i][j]   = ||r_i||^2 - 2 r_i . c_j
// then add ||c_j||^2 per lane (j = lane%16) and min-accumulate.
__global__ __launch_bounds__(128)
void chamfer_min_kernel(const float* __restrict__ pred,
                        const float* __restrict__ gt,
                        float* __restrict__ minOut, int N)
{
    const int lane    = threadIdx.x;                       // 0..31 (wave32)
    const int rowTile = blockIdx.x * blockDim.y + threadIdx.y;
    const int b       = blockIdx.y;
    const int pass    = blockIdx.z;

    const float* rows = (pass == 0) ? gt : pred;
    const float* cols = (pass == 0) ? pred : gt;
    rows += (size_t)b * N * 3;
    cols += (size_t)b * N * 3;

    const int half = lane >> 4;    // 0: lanes 0-15, 1: lanes 16-31
    const int l16  = lane & 15;

    // ---- Build A (16x4 f32): lane holds M=l16; VGPR v, half h -> K = v + 2h
    {
    }
    const int m = rowTile * 16 + l16;
    const float ax = rows[m * 3 + 0];
    const float ay = rows[m * 3 + 1];
    const float az = rows[m * 3 + 2];
    const float asq = ax * ax + ay * ay + az * az;
    v2f A;
    A.x = half ? az  : ax;   // K = 0 (h=0) / K = 2 (h=1)
    A.y = half ? asq : ay;   // K = 1 (h=0) / K = 3 (h=1)

    // Running min of squared distances in the C/D accumulator layout.
    v8f mn;
#pragma unroll
    for (int r = 0; r < 8; ++r) mn[r] = 3.0e38f;

    const int nTiles = N >> 4;
#pragma unroll 2
    for (int t = 0; t < nTiles; ++t) {
        const int n = t * 16 + l16;                 // column point index (N = l16)
        const float bx = cols[n * 3 + 0];
        const float by = cols[n * 3 + 1];
        const float bz = cols[n * 3 + 2];
        const float bsq = bx * bx + by * by + bz * bz;

        v2f Bm;                                     // B layout symmetric with A
        Bm.x = half ? (-2.0f * bz) : (-2.0f * bx);  // K = 0 / K = 2
        Bm.y = half ? 1.0f         : (-2.0f * by);  // K = 1 / K = 3

        v8f c = {};                                 // zero accumulator
        v8f d = __builtin_amdgcn_wmma_f32_16x16x4_f32(
            /*neg_a=*/false, A, /*neg_b=*/false, Bm,
            /*c_mod=*/(short)0, c, /*reuse_a=*/false, /*reuse_b=*/false);

#pragma unroll
        for (int r = 0; r < 8; ++r) {
            const float s = d[r] + bsq;             // + ||c_j||^2 (per-lane column)
            mn[r] = fminf(mn[r], s);
        }
    }

    // Cross-lane min within each 16-lane half (rows differ between halves).
#pragma unroll
    for (int off = 1; off < 16; off <<= 1) {
#pragma unroll
        for (int r = 0; r < 8; ++r) {
            const float o = __shfl_xor(mn[r], off, 32);
            mn[r] = fminf(mn[r], o);
        }
    }

    // Lane 0 holds mins for rows rowTile*16+0..7, lane 16 for +8..15.
    if (l16 == 0) {
        float* out = minOut + ((size_t)pass * gridDim.y + b) * N
                   + (size_t)rowTile * 16 + half * 8;
#pragma unroll
        for (int r = 0; r < 8; ++r)
            out[r] = sqrtf(fmaxf(mn[r], 0.0f));
    }
}

// Sum all 2*B*N per-point nearest distances, scale by 1/(B*N).
__global__ __launch_bounds__(256)
void chamfer_reduce_kernel(const float* __restrict__ mins, float* __restrict__ out,
                           int total, float scale)
{
    __shared__ float sdata[256];
    float s = 0.0f;
    for (int i = threadIdx.x; i < total; i += blockDim.x) s += mins[i];
    sdata[threadIdx.x] = s;
    __syncthreads();
    for (int stride = 128; stride > 0; stride >>= 1) {
        if ((int)threadIdx.x < stride) sdata[threadIdx.x] += sdata[threadIdx.x + stride];
        __syncthreads();
    }
    if (threadIdx.x == 0) out[0] = sdata[0] * scale;
}

extern "C" void kernel_launch(void* const* d_in, const int* in_sizes, int n_in,
                              void* d_out, int out_size, void* d_ws, size_t ws_size,
                              hipStream_t stream)
{
    const float* pred = (const float*)d_in[0];
    const float* gt   = (const float*)d_in[1];

    const int B = 2;
    const int N = in_sizes[0] / (B * 3);   // 8192

    float* mins = (float*)d_ws;            // [2][B][N] floats (fully overwritten)

    dim3 block(32, 4, 1);                  // 4 waves per workgroup, wave32
    dim3 grid((N / 16) / 4, B, 2);         // row tiles x batch x pass

    chamfer_min_kernel<<<grid, block, 0, stream>>>(pred, gt, mins, N);
    chamfer_reduce_kernel<<<1, 256, 0, stream>>>(mins, (float*)d_out,
                                                 2 * B * N, 1.0f / (float)(B * N));
}